// Spatial_Multi_Layer_74311524155898
// MI455X (gfx1250) — compile-verified
//
#include <hip/hip_runtime.h>

typedef __attribute__((ext_vector_type(2))) float v2f;
typedef __attribute__((ext_vector_type(4))) float v4f;
typedef __attribute__((ext_vector_type(8))) float v8f;
typedef __attribute__((ext_vector_type(4))) int   v4i;
typedef __attribute__((address_space(3))) v4i*    lds_v4i_p;   // 32-bit LDS pointer

#define N_   32
#define CIN  64
#define COUT 64
#define T_   512
#define V_   25
#define S_   3
#define R_   32
#define TT   16            // T-tile rows per block
#define COLS (TT * V_)     // 400 (t,v) columns, contiguous in x
#define XST  404           // LDS row stride (16B-aligned rows, bank-skewed)
#define VST  404

// workspace layout (float offsets)
#define WS_BN_SUM  0                 // 64
#define WS_BN_SQ   64                // 64
#define WS_BN_MU   128               // 64
#define WS_BN_RSIG 192               // 64
#define WS_XM      256               // N*CIN*V = 51200
#define WS_ATT     (256 + 51200)     // N*S*COUT*V*V = 3,840,000  (~15.6 MB total ws)

__global__ void k_zero(float* ws) {
  int i = threadIdx.x;
  if (i < 256) ws[i] = 0.f;
}

// xm[n,c,v] = mean_t x[n,c,t,v]
__global__ void k_xm(const float* __restrict__ x, float* __restrict__ ws) {
  int nc = blockIdx.x;            // n*CIN + c
  int v  = threadIdx.x;
  if (v >= V_) return;
  const float* p = x + (size_t)nc * (T_ * V_) + v;
  float acc = 0.f;
  for (int t = 0; t < T_; ++t) acc += p[t * V_];
  ws[WS_XM + nc * V_ + v] = acc * (1.0f / T_);
}

// att[n,s,o,u,v] = (sum_r w4[s,o,r]*relu(q[r,u]-k[r,v]) + b4[s,o])*alpha[s] + A[s,u,v]
__global__ void k_att(const float* __restrict__ A,
                      const float* __restrict__ w1, const float* __restrict__ b1,
                      const float* __restrict__ w2, const float* __restrict__ b2,
                      const float* __restrict__ w4, const float* __restrict__ b4,
                      const float* __restrict__ alpha,
                      float* __restrict__ ws) {
  int s = blockIdx.x, n = blockIdx.y, tid = threadIdx.x;
  __shared__ float xmS[CIN * V_];
  __shared__ float w1S[R_ * CIN], w2S[R_ * CIN];
  __shared__ float w4S[COUT * R_];
  __shared__ float AS[V_ * V_];
  __shared__ float b1S[R_], b2S[R_], b4S[COUT];
  __shared__ float qS[R_ * V_], kS[R_ * V_];
  const float* xm = ws + WS_XM + (size_t)n * CIN * V_;
  for (int i = tid; i < CIN * V_; i += 256) xmS[i] = xm[i];
  for (int i = tid; i < R_ * CIN; i += 256) {
    w1S[i] = w1[s * R_ * CIN + i];
    w2S[i] = w2[s * R_ * CIN + i];
  }
  for (int i = tid; i < COUT * R_; i += 256) w4S[i] = w4[s * COUT * R_ + i];
  for (int i = tid; i < V_ * V_; i += 256) AS[i] = A[s * V_ * V_ + i];
  if (tid < R_)   { b1S[tid] = b1[s * R_ + tid]; b2S[tid] = b2[s * R_ + tid]; }
  if (tid < COUT)   b4S[tid] = b4[s * COUT + tid];
  __syncthreads();
  for (int i = tid; i < R_ * V_; i += 256) {
    int r = i / V_, u = i % V_;
    float aq = 0.f, ak = 0.f;
    for (int c = 0; c < CIN; ++c) {
      float xv = xmS[c * V_ + u];
      aq += w1S[r * CIN + c] * xv;
      ak += w2S[r * CIN + c] * xv;
    }
    qS[i] = aq + b1S[r];
    kS[i] = ak + b2S[r];
  }
  __syncthreads();
  float al = alpha[s];
  float* attg = ws + WS_ATT + ((size_t)(n * S_ + s) * COUT) * (V_ * V_);
  for (int i = tid; i < COUT * V_ * V_; i += 256) {
    int o = i / (V_ * V_);
    int rem = i % (V_ * V_);
    int u = rem / V_, v = rem % V_;
    float acc = 0.f;
    for (int r = 0; r < R_; ++r) {
      float d = qS[r * V_ + u] - kS[r * V_ + v];
      acc += w4S[o * R_ + r] * fmaxf(d, 0.f);
    }
    attg[i] = (acc + b4S[o]) * al + AS[rem];
  }
}

// Fused conv3 + attention aggregation, all in f32 WMMA (16x16x4).
// grid (T/TT, N), 512 threads = 16 waves, dynamic LDS (~224 KB, 1 WG / WGP).
__global__ void __launch_bounds__(512) k_main(const float* __restrict__ x,
                                              const float* __restrict__ w3,
                                              const float* __restrict__ b3,
                                              float* __restrict__ y,
                                              float* __restrict__ ws) {
  extern __shared__ float sm[];
  float* xs   = sm;                      // [CIN][XST]  x tile, col = t_local*25 + v
  float* vbuf = xs + CIN * XST;          // [COUT][VST] v = w3.x + b3 for current s
  float* w3s  = vbuf + COUT * VST;       // [COUT][CIN]
  float* b3s  = w3s + COUT * CIN;        // [COUT]

  const int tchunk = blockIdx.x, n = blockIdx.y;
  const int t0   = tchunk * TT;
  const int tid  = threadIdx.x;
  const int lane = tid & 31, w = tid >> 5;       // 16 waves
  const int l16  = lane & 15;
  const int koff = (lane < 16) ? 0 : 2;          // f32 WMMA A/B K sub-lane split
  const int mofs = (lane < 16) ? 0 : 8;          // f32 WMMA D row split

  // x[n, c, t0:t0+16, 0:25] is contiguous per c (400 floats = 100 b128).
  const float* xg = x + ((size_t)n * CIN) * (T_ * V_) + (size_t)t0 * V_;
#if defined(__gfx1250__) && __has_builtin(__builtin_amdgcn_global_load_async_to_lds_b128)
  for (int i = tid; i < CIN * (COLS / 4); i += 512) {
    int c = i / (COLS / 4), j4 = i - c * (COLS / 4);
    const float* src = xg + (size_t)c * (T_ * V_) + 4 * j4;
    float* dst = &xs[c * XST + 4 * j4];
    __builtin_amdgcn_global_load_async_to_lds_b128(
        (v4i*)src,
        (lds_v4i_p)(unsigned int)(uintptr_t)dst,
        0, 0);
  }
#if __has_builtin(__builtin_amdgcn_s_wait_asynccnt)
  __builtin_amdgcn_s_wait_asynccnt(0);
#else
  asm volatile("s_wait_asynccnt 0" ::: "memory");
#endif
#else
  for (int i = tid; i < CIN * (COLS / 4); i += 512) {
    int c = i / (COLS / 4), j4 = i - c * (COLS / 4);
    *(v4f*)&xs[c * XST + 4 * j4] = *(const v4f*)(xg + (size_t)c * (T_ * V_) + 4 * j4);
  }
#endif

  v8f yacc[4][2];
  for (int a = 0; a < 4; ++a)
    for (int b = 0; b < 2; ++b)
      yacc[a][b] = (v8f){0.f, 0.f, 0.f, 0.f, 0.f, 0.f, 0.f, 0.f};

  for (int s = 0; s < S_; ++s) {
    __syncthreads();   // vbuf/w3s free to overwrite; xs copies complete
    for (int i = tid; i < COUT * CIN / 4; i += 512)
      *(v4f*)&w3s[4 * i] = *(const v4f*)(w3 + (size_t)s * COUT * CIN + 4 * i);
    if (tid < COUT) b3s[tid] = b3[s * COUT + tid];
    __syncthreads();

    // Stage A: vbuf[o, col] = sum_c w3[s,o,c] * xs[c, col] + b3[s,o]
    // D tiles: 4 o-tiles x 25 col-tiles; K = 64 in steps of 4.
    for (int tt = w; tt < 4 * 25; tt += 16) {
      int otile = tt & 3, ct = tt >> 2;
      int colb = ct * 16 + l16;
      v8f acc = (v8f){0.f, 0.f, 0.f, 0.f, 0.f, 0.f, 0.f, 0.f};
      for (int k4 = 0; k4 < CIN; k4 += 4) {
        v2f a, b;
        a.x = w3s[(otile * 16 + l16) * CIN + k4 + koff];
        a.y = w3s[(otile * 16 + l16) * CIN + k4 + koff + 1];
        b.x = xs[(k4 + koff) * XST + colb];
        b.y = xs[(k4 + koff + 1) * XST + colb];
        acc = __builtin_amdgcn_wmma_f32_16x16x4_f32(false, a, false, b,
                                                    (short)0, acc, false, false);
      }
      for (int j = 0; j < 8; ++j) {
        int o = otile * 16 + j + mofs;
        vbuf[o * VST + colb] = acc[j] + b3s[o];
      }
    }
    __syncthreads();

    // Stage B: y[t,u] += sum_v vbuf[o][t,v] * att[o][u,v]  (K=v padded 25->28)
    const float* attg = ws + WS_ATT + ((size_t)(n * S_ + s) * COUT) * (V_ * V_);
    for (int oi = 0; oi < 4; ++oi) {
      int o = w + oi * 16;
      const float* ao = attg + o * (V_ * V_);
      const float* vo = vbuf + o * VST;
      for (int ut = 0; ut < 2; ++ut) {
        int u = ut * 16 + l16;
        bool uok = (u < V_);
        v8f acc = yacc[oi][ut];
        for (int k4 = 0; k4 < 28; k4 += 4) {
          int v0 = k4 + koff;
          v2f a, b;
          a.x = (v0 < V_)     ? vo[l16 * V_ + v0]     : 0.f;
          a.y = (v0 + 1 < V_) ? vo[l16 * V_ + v0 + 1] : 0.f;
          b.x = (uok && v0 < V_)     ? ao[u * V_ + v0]     : 0.f;
          b.y = (uok && v0 + 1 < V_) ? ao[u * V_ + v0 + 1] : 0.f;
          acc = __builtin_amdgcn_wmma_f32_16x16x4_f32(false, a, false, b,
                                                      (short)0, acc, false, false);
        }
        yacc[oi][ut] = acc;
      }
    }
  }

  // Write y and accumulate BN sums per channel o.
  for (int oi = 0; oi < 4; ++oi) {
    int o = w + oi * 16;
    float lsum = 0.f, lsq = 0.f;
    for (int ut = 0; ut < 2; ++ut) {
      int u = ut * 16 + l16;
      if (u < V_) {
        for (int j = 0; j < 8; ++j) {
          int tl = j + mofs;
          float val = yacc[oi][ut][j];
          y[(((size_t)n * COUT + o) * T_ + (t0 + tl)) * V_ + u] = val;
          lsum += val;
          lsq += val * val;
        }
      }
    }
    for (int m = 16; m >= 1; m >>= 1) {
      lsum += __shfl_xor(lsum, m, 32);
      lsq  += __shfl_xor(lsq,  m, 32);
    }
    if (lane == 0) {
      atomicAdd(&ws[WS_BN_SUM + o], lsum);
      atomicAdd(&ws[WS_BN_SQ + o], lsq);
    }
  }
}

__global__ void k_bnfin(float* ws) {
  int o = threadIdx.x;
  if (o < COUT) {
    const float cnt = (float)N_ * (float)T_ * (float)V_;
    float mu  = ws[WS_BN_SUM + o] / cnt;
    float var = ws[WS_BN_SQ + o] / cnt - mu * mu;
    ws[WS_BN_MU + o]   = mu;
    ws[WS_BN_RSIG + o] = rsqrtf(var + 1e-5f);
  }
}

// out = relu(y) + (y - mu)*rsig*gamma + beta + x   (in place on d_out, vec4)
__global__ void k_final(const float* __restrict__ x,
                        const float* __restrict__ gamma,
                        const float* __restrict__ beta,
                        const float* __restrict__ ws,
                        float* __restrict__ out, int total4) {
  int i4 = blockIdx.x * 256 + threadIdx.x;
  if (i4 >= total4) return;
  int o = (i4 / (T_ * V_ / 4)) & (COUT - 1);   // T*V divisible by 4 -> o uniform in vec4
  v4f yv = ((const v4f*)out)[i4];
  v4f xv = ((const v4f*)x)[i4];
  float mu = ws[WS_BN_MU + o], rs = ws[WS_BN_RSIG + o];
  float g = gamma[o], b = beta[o];
  v4f r;
  for (int j = 0; j < 4; ++j)
    r[j] = fmaxf(yv[j], 0.f) + (yv[j] - mu) * rs * g + b + xv[j];
  ((v4f*)out)[i4] = r;
}

extern "C" void kernel_launch(void* const* d_in, const int* in_sizes, int n_in,
                              void* d_out, int out_size, void* d_ws, size_t ws_size,
                              hipStream_t stream) {
  const float* x     = (const float*)d_in[0];
  const float* A     = (const float*)d_in[1];
  const float* w1    = (const float*)d_in[2];
  const float* b1    = (const float*)d_in[3];
  const float* w2    = (const float*)d_in[4];
  const float* b2    = (const float*)d_in[5];
  const float* w3    = (const float*)d_in[6];
  const float* b3    = (const float*)d_in[7];
  const float* w4    = (const float*)d_in[8];
  const float* b4    = (const float*)d_in[9];
  const float* alpha = (const float*)d_in[10];
  const float* gamma = (const float*)d_in[11];
  const float* beta  = (const float*)d_in[12];
  float* out = (float*)d_out;
  float* ws  = (float*)d_ws;

  k_zero<<<1, 256, 0, stream>>>(ws);
  k_xm<<<N_ * CIN, 32, 0, stream>>>(x, ws);
  k_att<<<dim3(S_, N_), 256, 0, stream>>>(A, w1, b1, w2, b2, w4, b4, alpha, ws);

  size_t lds_bytes = (size_t)(CIN * XST + COUT * VST + COUT * CIN + COUT) * sizeof(float);
  k_main<<<dim3(T_ / TT, N_), 512, lds_bytes, stream>>>(x, w3, b3, out, ws);

  k_bnfin<<<1, 64, 0, stream>>>(ws);
  int total4 = N_ * COUT * T_ * V_ / 4;
  k_final<<<(total4 + 255) / 256, 256, 0, stream>>>(x, gamma, beta, ws, out, total4);
}